// ProbAttention_8796093022780
// MI455X (gfx1250) — compile-verified
//
#include <hip/hip_runtime.h>
#include <math.h>

// Problem constants (fixed by the reference setup)
constexpr int B = 4, L = 4096, H = 8, D = 64;
constexpr int SK = 45;          // sample_k
constexpr int U  = 45;          // top-u queries
constexpr int UPAD = 48;        // padded to 3 WMMA m-tiles
constexpr int CHUNK = 128;      // keys per flash chunk (8 n-tiles)
constexpr int SPLIT = 8;        // key-range splits per (b,h) -> 256 WGs
constexpr int SEGT = 4;         // threads per row for softmax partials
constexpr int SEGW = CHUNK / SEGT;

typedef __attribute__((ext_vector_type(2))) float v2f;
typedef __attribute__((ext_vector_type(4))) float v4f;
typedef __attribute__((ext_vector_type(8))) float v8f;

// ---------------------------------------------------------------------------
// K1: mean of V over the key dimension, per (b,h,d)
// ---------------------------------------------------------------------------
__global__ __launch_bounds__(256)
void meanv_kernel(const float* __restrict__ V, float* __restrict__ mean) {
  int bh = blockIdx.x, b = bh / H, h = bh % H;
  int tid = threadIdx.x;
  int d = tid & 63, part = tid >> 6;          // 4 partial sums per d
  float s = 0.f;
  for (int l = part; l < L; l += 4)
    s += V[((size_t)(b * L + l) * H + h) * D + d];
  __shared__ float red[4][64];
  red[part][d] = s;
  __syncthreads();
  if (part == 0) {
    float m = (red[0][d] + red[1][d] + red[2][d] + red[3][d]) * (1.0f / (float)L);
    mean[bh * 64 + d] = m;
  }
}

// ---------------------------------------------------------------------------
// K2: broadcast the mean into the whole output; 128-bit non-temporal stores
// ---------------------------------------------------------------------------
__global__ __launch_bounds__(256)
void fill_kernel(const float* __restrict__ mean, float* __restrict__ out) {
  size_t i = (size_t)blockIdx.x * 256 + threadIdx.x;   // over float4 elements
  size_t total4 = (size_t)B * L * H * (D / 4);
  if (i >= total4) return;
  int d4 = (int)(i & 15);
  size_t t = i >> 4;                // (b*L + l)*H + h
  int h = (int)(t % H);
  int b = (int)((t / H) / L);
  v4f m = ((const v4f*)mean)[(b * H + h) * 16 + d4];
  __builtin_nontemporal_store(m, (v4f*)out + i);
}

// ---------------------------------------------------------------------------
// K3: sparsity measure M[b,h,l] = max_s(q.k_s) - sum_s(q.k_s)/L
// ---------------------------------------------------------------------------
__global__ __launch_bounds__(256)
void mscore_kernel(const float* __restrict__ Q, const float* __restrict__ K,
                   const long long* __restrict__ isamp, float* __restrict__ M) {
  int gid = blockIdx.x * 256 + threadIdx.x;   // b*H*L + h*L + l
  if (gid >= B * H * L) return;
  int l = gid % L;
  int h = (gid / L) % H;
  int b = gid / (L * H);
  const float4* qr = (const float4*)&Q[((size_t)(b * L + l) * H + h) * D];
  float4 q[16];
#pragma unroll
  for (int i = 0; i < 16; i++) q[i] = qr[i];
  float mx = -__builtin_inff(), sm = 0.f;
  int idx = (int)isamp[(size_t)l * SK];
  for (int s = 0; s < SK; s++) {
    const float4* kr = (const float4*)&K[((size_t)(b * L + idx) * H + h) * D];
    if (s + 1 < SK) {               // prefetch next gathered K row
      idx = (int)isamp[(size_t)l * SK + s + 1];
      __builtin_prefetch(&K[((size_t)(b * L + idx) * H + h) * D], 0, 0);
    }
    float dot = 0.f;
#pragma unroll
    for (int i = 0; i < 16; i++) {
      float4 kv = kr[i];
      dot += q[i].x * kv.x + q[i].y * kv.y + q[i].z * kv.z + q[i].w * kv.w;
    }
    mx = fmaxf(mx, dot);
    sm += dot;
  }
  M[gid] = mx - sm * (1.0f / (float)L);
}

// ---------------------------------------------------------------------------
// K4: top-U indices of M per (b,h), iterative argmax (ties -> lower index)
// ---------------------------------------------------------------------------
__global__ __launch_bounds__(256)
void topk_kernel(const float* __restrict__ M, int* __restrict__ topk) {
  int bh = blockIdx.x, tid = threadIdx.x;
  __shared__ float vals[L];          // 16 KB
  __shared__ float rv[256];
  __shared__ int   ri[256];
  for (int i = tid; i < L; i += 256) vals[i] = M[(size_t)bh * L + i];
  __syncthreads();
  for (int it = 0; it < U; it++) {
    float bv = -__builtin_inff();
    int bi = L;
    for (int i = tid; i < L; i += 256) {
      float v = vals[i];
      if (v > bv) { bv = v; bi = i; }    // ascending scan -> ties keep lower i
    }
    rv[tid] = bv; ri[tid] = bi;
    __syncthreads();
    for (int off = 128; off > 0; off >>= 1) {
      if (tid < off) {
        float v2 = rv[tid + off]; int i2 = ri[tid + off];
        if (v2 > rv[tid] || (v2 == rv[tid] && i2 < ri[tid])) { rv[tid] = v2; ri[tid] = i2; }
      }
      __syncthreads();
    }
    if (tid == 0) { topk[bh * U + it] = ri[0]; vals[ri[0]] = -__builtin_inff(); }
    __syncthreads();
  }
}

// ---------------------------------------------------------------------------
// K5: flash-attention partial over one key range, WMMA fp32 matrix pipe.
//     blockIdx.x = bh*SPLIT + s ; writes (m, l, acc) partials to workspace.
// ---------------------------------------------------------------------------
__global__ __launch_bounds__(256)
void attn_partial_kernel(const float* __restrict__ Q, const float* __restrict__ Km,
                         const float* __restrict__ V, const int* __restrict__ topk,
                         float* __restrict__ mpart, float* __restrict__ lpart,
                         float* __restrict__ accpart) {
  int blk = blockIdx.x;
  int bh = blk / SPLIT, sp = blk % SPLIT;
  int b = bh / H, h = bh % H;
  int tid = threadIdx.x;
  int lane = tid & 31;
  const int wid = __builtin_amdgcn_readfirstlane(tid >> 5);  // scalar wave id
  const int half = lane >> 4;        // 0: lanes 0-15, 1: lanes 16-31
  const int ln   = lane & 15;

  __shared__ float Qs[UPAD * D];        // 12 KB, pre-scaled queries
  __shared__ float Ss[UPAD * CHUNK];    // 24 KB, score / prob chunk
  __shared__ float acc[UPAD * D];       // 12 KB, running output
  __shared__ float mrow[UPAD], lrow[UPAD], corr[UPAD];
  __shared__ float pmax[UPAD * SEGT], psum[UPAD * SEGT];
  __shared__ int   qidx[UPAD];

  for (int i = tid; i < UPAD; i += 256) {
    qidx[i] = (i < U) ? topk[bh * U + i] : 0;
    mrow[i] = -__builtin_inff();
    lrow[i] = 0.f;
  }
  __syncthreads();
  for (int i = tid; i < UPAD * D; i += 256) {
    int r = i / D, d = i % D;
    Qs[i]  = (r < U) ? Q[((size_t)(b * L + qidx[r]) * H + h) * D + d] * 0.125f : 0.f;
    acc[i] = 0.f;
  }
  __syncthreads();

  const int kbeg = sp * (L / SPLIT);
  const int kend = kbeg + (L / SPLIT);
  for (int cb = kbeg; cb < kend; cb += CHUNK) {
    // prefetch next chunk's K / V rows
    if (cb + CHUNK < kend && tid < CHUNK) {
      __builtin_prefetch(&Km[((size_t)(b * L + cb + CHUNK + tid) * H + h) * D], 0, 0);
      __builtin_prefetch(&V [((size_t)(b * L + cb + CHUNK + tid) * H + h) * D], 0, 0);
    }
    // ---- Phase 1: S = Qs @ K_chunk^T  (3 x 8 = 24 tile jobs over 8 waves) ----
    for (int job = wid; job < (UPAD / 16) * (CHUNK / 16); job += 8) {
      int mt = job / (CHUNK / 16), nt = job % (CHUNK / 16);
      int m_base = mt * 16, n_base = nt * 16;
      v8f c = {};
      const float* krow = &Km[((size_t)(b * L + cb + n_base + ln) * H + h) * D];
#pragma unroll
      for (int kk = 0; kk < D; kk += 4) {
        // A (16x4 fp32): lanes 0-15 hold K=kk..kk+1, lanes 16-31 hold kk+2..kk+3
        float2 av = *(const float2*)&Qs[(m_base + ln) * D + kk + 2 * half];
        v2f a; a.x = av.x; a.y = av.y;
        // B (4x16 fp32) = K^T: element (krow_d, n) = K[key=n][d=kk+krow_d]
        float2 bv = *(const float2*)&krow[kk + 2 * half];
        v2f bb; bb.x = bv.x; bb.y = bv.y;
        c = __builtin_amdgcn_wmma_f32_16x16x4_f32(false, a, false, bb,
                                                  (short)0, c, false, false);
      }
      // C layout: VGPR r -> row (r + 8*half), col ln
#pragma unroll
      for (int r = 0; r < 8; r++)
        Ss[(m_base + half * 8 + r) * CHUNK + n_base + ln] = c[r];
    }
    __syncthreads();

    // ---- Phase 2a: 4-way partial row max ----
    if (tid < UPAD * SEGT) {
      int r = tid / SEGT, sg = tid % SEGT;
      const float* row = &Ss[r * CHUNK + sg * SEGW];
      float cm = -__builtin_inff();
      for (int j = 0; j < SEGW; j++) cm = fmaxf(cm, row[j]);
      pmax[tid] = cm;
    }
    __syncthreads();
    if (tid < UPAD) {
      float cm = fmaxf(fmaxf(pmax[tid * SEGT + 0], pmax[tid * SEGT + 1]),
                       fmaxf(pmax[tid * SEGT + 2], pmax[tid * SEGT + 3]));
      float mn = fmaxf(mrow[tid], cm);
      float cr = __expf(mrow[tid] - mn);     // 0 on the first chunk
      mrow[tid] = mn;
      corr[tid] = cr;
      lrow[tid] *= cr;
    }
    __syncthreads();
    // ---- Phase 2b: exp + partial sums (threads 0..191) | acc rescale (192..255)
    if (tid < UPAD * SEGT) {
      int r = tid / SEGT, sg = tid % SEGT;
      float* row = &Ss[r * CHUNK + sg * SEGW];
      float mn = mrow[r], s = 0.f;
      for (int j = 0; j < SEGW; j++) { float p = __expf(row[j] - mn); row[j] = p; s += p; }
      psum[tid] = s;
    } else {
      for (int i = tid - UPAD * SEGT; i < UPAD * D; i += 256 - UPAD * SEGT)
        acc[i] *= corr[i / D];
    }
    __syncthreads();
    if (tid < UPAD)
      lrow[tid] += psum[tid * SEGT + 0] + psum[tid * SEGT + 1] +
                   psum[tid * SEGT + 2] + psum[tid * SEGT + 3];
    __syncthreads();

    // ---- Phase 3: acc += P @ V_chunk (3 x 4 = 12 tile jobs over 8 waves) ----
    for (int job = wid; job < (UPAD / 16) * (D / 16); job += 8) {
      int mt = job / (D / 16), nt = job % (D / 16);
      int m_base = mt * 16, n_base = nt * 16;
      v8f c;
#pragma unroll
      for (int r = 0; r < 8; r++)
        c[r] = acc[(m_base + half * 8 + r) * D + n_base + ln];
      for (int kk = 0; kk < CHUNK; kk += 4) {
        float2 av = *(const float2*)&Ss[(m_base + ln) * CHUNK + kk + 2 * half];
        v2f a; a.x = av.x; a.y = av.y;
        // B (4x16) = V rows kk..kk+3, cols n_base..n_base+15
        size_t vbase = ((size_t)(b * L + cb + kk + 2 * half) * H + h) * D + n_base + ln;
        v2f bb;
        bb.x = V[vbase];
        bb.y = V[vbase + (size_t)H * D];
        c = __builtin_amdgcn_wmma_f32_16x16x4_f32(false, a, false, bb,
                                                  (short)0, c, false, false);
      }
#pragma unroll
      for (int r = 0; r < 8; r++)
        acc[(m_base + half * 8 + r) * D + n_base + ln] = c[r];
    }
    __syncthreads();
  }

  // ---- write partials ----
  for (int i = tid; i < UPAD; i += 256) {
    mpart[(size_t)blk * UPAD + i] = mrow[i];
    lpart[(size_t)blk * UPAD + i] = lrow[i];
  }
  for (int i = tid; i < UPAD * D; i += 256)
    accpart[(size_t)blk * UPAD * D + i] = acc[i];
}

// ---------------------------------------------------------------------------
// K6: merge the SPLIT flash partials per (b,h) and scatter into the output
// ---------------------------------------------------------------------------
__global__ __launch_bounds__(256)
void combine_kernel(const float* __restrict__ mpart, const float* __restrict__ lpart,
                    const float* __restrict__ accpart, const int* __restrict__ topk,
                    float* __restrict__ out) {
  int bh = blockIdx.x, b = bh / H, h = bh % H;
  int tid = threadIdx.x;
  __shared__ float w[SPLIT][UPAD];
  __shared__ float linv[UPAD];
  __shared__ int   qidx[U];
  if (tid < UPAD) {
    float mstar = -__builtin_inff();
    for (int s = 0; s < SPLIT; s++)
      mstar = fmaxf(mstar, mpart[(size_t)(bh * SPLIT + s) * UPAD + tid]);
    float lsum = 0.f;
    for (int s = 0; s < SPLIT; s++) {
      float ww = __expf(mpart[(size_t)(bh * SPLIT + s) * UPAD + tid] - mstar);
      w[s][tid] = ww;
      lsum += lpart[(size_t)(bh * SPLIT + s) * UPAD + tid] * ww;
    }
    linv[tid] = 1.f / lsum;
  }
  if (tid < U) qidx[tid] = topk[bh * U + tid];
  __syncthreads();
  for (int i = tid; i < U * D; i += 256) {
    int r = i / D, d = i % D;
    float a = 0.f;
    for (int s = 0; s < SPLIT; s++)
      a += accpart[((size_t)(bh * SPLIT + s) * UPAD + r) * D + d] * w[s][r];
    out[((size_t)(b * L + qidx[r]) * H + h) * D + d] = a * linv[r];
  }
}

// ---------------------------------------------------------------------------
extern "C" void kernel_launch(void* const* d_in, const int* in_sizes, int n_in,
                              void* d_out, int out_size, void* d_ws, size_t ws_size,
                              hipStream_t stream) {
  const float*     Q     = (const float*)d_in[0];
  const float*     K     = (const float*)d_in[1];
  const float*     V     = (const float*)d_in[2];
  const long long* isamp = (const long long*)d_in[3];   // int64 indices
  float* out = (float*)d_out;

  float* ws      = (float*)d_ws;
  float* mean    = ws;                                   // B*H*64
  float* M       = mean + B * H * 64;                    // B*H*L
  int*   topk    = (int*)(M + (size_t)B * H * L);        // B*H*U
  float* mpart   = (float*)(topk + B * H * U);           // B*H*SPLIT*UPAD
  float* lpart   = mpart + (size_t)B * H * SPLIT * UPAD;
  float* accpart = lpart + (size_t)B * H * SPLIT * UPAD; // B*H*SPLIT*UPAD*D

  meanv_kernel<<<B * H, 256, 0, stream>>>(V, mean);
  fill_kernel<<<(int)(((size_t)B * L * H * (D / 4) + 255) / 256), 256, 0, stream>>>(mean, out);
  mscore_kernel<<<(B * H * L + 255) / 256, 256, 0, stream>>>(Q, K, isamp, M);
  topk_kernel<<<B * H, 256, 0, stream>>>(M, topk);
  attn_partial_kernel<<<B * H * SPLIT, 256, 0, stream>>>(Q, K, V, topk,
                                                         mpart, lpart, accpart);
  combine_kernel<<<B * H, 256, 0, stream>>>(mpart, lpart, accpart, topk, out);
}